// SelfAttentionBlock_33595234189476
// MI455X (gfx1250) — compile-verified
//
#include <hip/hip_runtime.h>

typedef __attribute__((ext_vector_type(2))) float v2f;
typedef __attribute__((ext_vector_type(8))) float v8f;

#define QK_SCALE 0.25f   // HD=16 -> 16^-0.5

// ---------------------------------------------------------------------------
// Order-preserving int encoding of float for atomicMax-based segment max
// ---------------------------------------------------------------------------
__device__ __forceinline__ int ordFloat(float f) {
  int b = __float_as_int(f);
  return b >= 0 ? b : (b ^ 0x7fffffff);
}
__device__ __forceinline__ float deordFloat(int e) {
  int b = e >= 0 ? e : (e ^ 0x7fffffff);
  return __int_as_float(b);
}

// ---------------------------------------------------------------------------
// Kernel 0: init d_out = 0, segsum = 0, segmax = -inf (encoded INT_MIN)
// ---------------------------------------------------------------------------
__global__ void init_kernel(float* __restrict__ out, float* __restrict__ segsum,
                            int* __restrict__ segmax, long long n_out, long long nh)
{
  long long i = (long long)blockIdx.x * blockDim.x + threadIdx.x;
  if (i < n_out) out[i] = 0.0f;
  if (i < nh) {
    segsum[i] = 0.0f;
    segmax[i] = (int)0x80000000;  // INT_MIN: below every encoded float
  }
}

// ---------------------------------------------------------------------------
// Kernel 1: qkv = x @ W_qkv^T + b, fp32 WMMA (V_WMMA_F32_16X16X4_F32)
// One wave computes a 16-row x 192-col tile of qkv.
//   A (16x4 f32): lane<16 -> x[row0+lane][4k+0..1], lane>=16 -> [4k+2..3]
//   B (4x16 f32): lane<16 -> W[col][4k+0..1],       lane>=16 -> [4k+2..3]
//   C/D (16x16):  vgpr r = row (lane<16 ? r : 8+r), col = lane&15
// ---------------------------------------------------------------------------
__global__ void __launch_bounds__(256) qkv_gemm_kernel(
    const float* __restrict__ x, const float* __restrict__ W,
    const float* __restrict__ b, float* __restrict__ qkv, int nTiles)
{
  const int wave = threadIdx.x >> 5;
  const int lane = threadIdx.x & 31;
  const int tile = blockIdx.x * (blockDim.x >> 5) + wave;
  if (tile >= nTiles) return;           // wave-uniform: EXEC all-ones below

  const int row0  = tile << 4;
  const int m     = lane & 15;
  const int khalf = (lane >> 4) << 1;   // 0 or 2

  // Preload all 16 A fragments (x tile 16x64, K consumed 4 per WMMA)
  v2f a[16];
  const float* xrow = x + (size_t)(row0 + m) * 64 + khalf;
#pragma unroll
  for (int kk = 0; kk < 16; ++kk)
    a[kk] = *(const v2f*)(xrow + kk * 4);

  const int n     = lane & 15;
  const int rbase = row0 + ((lane >> 4) << 3);

#pragma unroll 1
  for (int c = 0; c < 12; ++c) {
    const int col = (c << 4) + n;
    const float bias = b[col];
    v8f acc = {bias, bias, bias, bias, bias, bias, bias, bias};
    const float* wrow = W + (size_t)col * 64 + khalf;  // B[k][col] = W[col][k]
#pragma unroll
    for (int kk = 0; kk < 16; ++kk) {
      v2f bb = *(const v2f*)(wrow + kk * 4);
      acc = __builtin_amdgcn_wmma_f32_16x16x4_f32(
          false, a[kk], false, bb, (short)0, acc, false, false);
    }
#pragma unroll
    for (int r = 0; r < 8; ++r)
      qkv[(size_t)(rbase + r) * 192 + col] = acc[r];
  }
}

// ---------------------------------------------------------------------------
// Kernel 2: per (edge, head): compat = QK_SCALE * dot(q[s,h], k[t,h]);
//           segment max into segmax[s*4+h] (ordered-int atomicMax).
// qkv row layout: [0:64)=q, [64:128)=k, [128:192)=v; head h -> 16 floats
// ---------------------------------------------------------------------------
__global__ void __launch_bounds__(256) edge_compat_kernel(
    const float* __restrict__ qkv, const long long* __restrict__ ei,
    float* __restrict__ compat, int* __restrict__ segmax, long long E)
{
  long long gid = (long long)blockIdx.x * blockDim.x + threadIdx.x;
  if (gid >= E * 4) return;
  long long e = gid >> 2;
  int h = (int)(gid & 3);
  long long s = ei[e];
  long long t = ei[E + e];

  const float4* qp = (const float4*)(qkv + s * 192 + h * 16);
  const float4* kp = (const float4*)(qkv + t * 192 + 64 + h * 16);
  float acc = 0.0f;
#pragma unroll
  for (int i = 0; i < 4; ++i) {
    float4 q = qp[i];
    float4 k = kp[i];
    acc += q.x * k.x + q.y * k.y + q.z * k.z + q.w * k.w;
  }
  acc *= QK_SCALE;
  compat[gid] = acc;
  atomicMax(&segmax[s * 4 + h], ordFloat(acc));
}

// ---------------------------------------------------------------------------
// Kernel 3: ex = exp(compat - segmax[s,h]); segment sum into segsum[s,h]
// ---------------------------------------------------------------------------
__global__ void __launch_bounds__(256) edge_exp_kernel(
    const long long* __restrict__ ei, float* __restrict__ compat,
    const int* __restrict__ segmax, float* __restrict__ segsum, long long E)
{
  long long gid = (long long)blockIdx.x * blockDim.x + threadIdx.x;
  if (gid >= E * 4) return;
  long long e = gid >> 2;
  int h = (int)(gid & 3);
  long long s = ei[e];

  float c  = compat[gid];
  float mf = deordFloat(segmax[s * 4 + h]);
  float ex = __expf(c - mf);
  compat[gid] = ex;                      // reuse buffer: now holds ex
  unsafeAtomicAdd(&segsum[s * 4 + h], ex);
}

// ---------------------------------------------------------------------------
// Kernel 4: attn = ex / (segsum + 1e-16); out[s] += attn * v[t] (f32 atomics)
// ---------------------------------------------------------------------------
__global__ void __launch_bounds__(256) edge_scatter_kernel(
    const float* __restrict__ qkv, const long long* __restrict__ ei,
    const float* __restrict__ ex_buf, const float* __restrict__ segsum,
    float* __restrict__ out, long long E)
{
  long long gid = (long long)blockIdx.x * blockDim.x + threadIdx.x;
  if (gid >= E * 4) return;
  long long e = gid >> 2;
  int h = (int)(gid & 3);
  long long s = ei[e];
  long long t = ei[E + e];

  float attn = ex_buf[gid] / (segsum[s * 4 + h] + 1e-16f);
  const float4* vp = (const float4*)(qkv + t * 192 + 128 + h * 16);
  float* op = out + s * 64 + h * 16;
#pragma unroll
  for (int i = 0; i < 4; ++i) {
    float4 v = vp[i];
    unsafeAtomicAdd(op + i * 4 + 0, v.x * attn);
    unsafeAtomicAdd(op + i * 4 + 1, v.y * attn);
    unsafeAtomicAdd(op + i * 4 + 2, v.z * attn);
    unsafeAtomicAdd(op + i * 4 + 3, v.w * attn);
  }
}

// ---------------------------------------------------------------------------
// Host launcher
// ---------------------------------------------------------------------------
extern "C" void kernel_launch(void* const* d_in, const int* in_sizes, int n_in,
                              void* d_out, int out_size, void* d_ws, size_t ws_size,
                              hipStream_t stream)
{
  const float*     x  = (const float*)d_in[0];       // (N, 64)
  const long long* ei = (const long long*)d_in[1];   // (2, E) int64
  const float*     W  = (const float*)d_in[2];       // (192, 64)
  const float*     b  = (const float*)d_in[3];       // (192,)
  float* out = (float*)d_out;                        // (N, 64)

  const long long N = in_sizes[0] / 64;
  const long long E = in_sizes[1] / 2;

  // Workspace layout (all from d_ws):
  float* qkv    = (float*)d_ws;                      // N*192 f32
  float* compat = qkv + (size_t)N * 192;             // E*4   f32 (compat, then ex)
  float* segsum = compat + (size_t)E * 4;            // N*4   f32
  int*   segmax = (int*)(segsum + (size_t)N * 4);    // N*4   i32

  // 0) init out / segsum / segmax
  {
    long long n_out = N * 64, nh = N * 4;
    long long total = n_out;  // n_out > nh
    dim3 grid((unsigned)((total + 255) / 256));
    init_kernel<<<grid, 256, 0, stream>>>(out, segsum, segmax, n_out, nh);
  }
  // 1) qkv projection via fp32 WMMA
  {
    int nTiles = (int)((N + 15) / 16);
    int wavesPerBlock = 8;  // 256 threads
    dim3 grid((unsigned)((nTiles + wavesPerBlock - 1) / wavesPerBlock));
    qkv_gemm_kernel<<<grid, 256, 0, stream>>>(x, W, b, qkv, nTiles);
  }
  // 2..4) edge passes, one thread per (edge, head)
  {
    long long total = E * 4;
    dim3 grid((unsigned)((total + 255) / 256));
    edge_compat_kernel <<<grid, 256, 0, stream>>>(qkv, ei, compat, segmax, E);
    edge_exp_kernel    <<<grid, 256, 0, stream>>>(ei, compat, segmax, segsum, E);
    edge_scatter_kernel<<<grid, 256, 0, stream>>>(qkv, ei, compat, segsum, out, E);
  }
}